// YoloLoss_13443247636702
// MI455X (gfx1250) — compile-verified
//
#include <hip/hip_runtime.h>

typedef float v2f __attribute__((ext_vector_type(2)));
typedef float v8f __attribute__((ext_vector_type(8)));

#define S_DIM 7
#define C_DIM 20
#define CELL_F 30                      // floats per cell
#define NCELLS (65536 * S_DIM * S_DIM) // 3,211,264
#define THREADS 256
#define BLOCKS 2048

// ---------------------------------------------------------------------------
// IoU exactly as the reference (midpoint boxes, abs areas, 1e-6 eps).
// ---------------------------------------------------------------------------
__device__ __forceinline__ float iou_f(float ax, float ay, float aw, float ah,
                                       float bx, float by, float bw, float bh) {
    float ax1 = ax - aw * 0.5f, ay1 = ay - ah * 0.5f;
    float ax2 = ax + aw * 0.5f, ay2 = ay + ah * 0.5f;
    float bx1 = bx - bw * 0.5f, by1 = by - bh * 0.5f;
    float bx2 = bx + bw * 0.5f, by2 = by + bh * 0.5f;
    float x1 = fmaxf(ax1, bx1), y1 = fmaxf(ay1, by1);
    float x2 = fminf(ax2, bx2), y2 = fminf(ay2, by2);
    float inter  = fmaxf(x2 - x1, 0.0f) * fmaxf(y2 - y1, 0.0f);
    float area_a = fabsf((ax2 - ax1) * (ay2 - ay1));
    float area_b = fabsf((bx2 - bx1) * (by2 - by1));
    return inter / (area_a + area_b - inter + 1e-6f);
}

// ---------------------------------------------------------------------------
// Deterministic 256-thread block reduction. Final 256->1 stage runs on wave 0
// through V_WMMA_F32_16X16X4_F32: A(16x4) holds 64 pre-summed partials,
// B(4x16)=ones, so D rows are row-sums; one D column summed across the two
// lane halves (lane L and L+16) is the block total.
// Valid on thread 0's return value.
// ---------------------------------------------------------------------------
__device__ __forceinline__ float block_reduce_256(float v, float* sdata) {
    sdata[threadIdx.x] = v;
    __syncthreads();
    float total = 0.0f;
    if (threadIdx.x < 32) {
        const int L = threadIdx.x;
        // Fold 256 -> 64 values (2 A-slots per lane), exact f32 adds.
        float a0 = (sdata[L]       + sdata[L + 64])  + (sdata[L + 128] + sdata[L + 192]);
        float a1 = (sdata[L + 32]  + sdata[L + 96])  + (sdata[L + 160] + sdata[L + 224]);
#if __has_builtin(__builtin_amdgcn_wmma_f32_16x16x4_f32)
        v2f a; a.x = a0; a.y = a1;
        v2f b; b.x = 1.0f; b.y = 1.0f;        // B = ones(4x16)
        v8f c = {};
        // D[m][n] = sum_k A[m][k] : every D element is a row-sum of A.
        c = __builtin_amdgcn_wmma_f32_16x16x4_f32(
                /*neg_a=*/false, a, /*neg_b=*/false, b,
                /*c_mod=*/(short)0, c, /*reuse_a=*/false, /*reuse_b=*/false);
        // Lane n (0..15) holds D[0..7][n]; lane 16+n holds D[8..15][n].
        float cs = ((c[0] + c[1]) + (c[2] + c[3])) + ((c[4] + c[5]) + (c[6] + c[7]));
        total = cs + __shfl_xor(cs, 16, 32);   // add the two M-halves
#else
        // Fallback: plain wave32 shuffle tree over the 64 folded values.
        float s = a0 + a1;
        #pragma unroll
        for (int off = 16; off > 0; off >>= 1) s += __shfl_down(s, off, 32);
        total = s;
#endif
    }
    return total;
}

// ---------------------------------------------------------------------------
// Kernel 1: grid-stride over cells; streaming NT float2 loads; per-block
// partial sums into d_ws.
// ---------------------------------------------------------------------------
__global__ void __launch_bounds__(THREADS)
yolo_loss_partial(const float* __restrict__ pred, const float* __restrict__ tgt,
                  float* __restrict__ partial, int ncells) {
    float acc = 0.0f;
    const int stride = gridDim.x * blockDim.x;
    for (int cell = blockIdx.x * blockDim.x + threadIdx.x; cell < ncells; cell += stride) {
        const v2f* p2 = (const v2f*)(pred + (size_t)cell * CELL_F);
        const v2f* t2 = (const v2f*)(tgt  + (size_t)cell * CELL_F);

        float p[30];
        float t[26];
        #pragma unroll
        for (int i = 0; i < 15; ++i) {           // 30 pred floats, b64 NT loads
            v2f x = __builtin_nontemporal_load(p2 + i);
            p[2 * i] = x.x; p[2 * i + 1] = x.y;
        }
        #pragma unroll
        for (int i = 0; i < 13; ++i) {           // target floats 0..25 (need 0..24)
            v2f x = __builtin_nontemporal_load(t2 + i);
            t[2 * i] = x.x; t[2 * i + 1] = x.y;
        }

        // bestbox from the two IoUs (target box is ch 21..24 for both)
        float iou1 = iou_f(p[21], p[22], p[23], p[24], t[21], t[22], t[23], t[24]);
        float iou2 = iou_f(p[26], p[27], p[28], p[29], t[21], t[22], t[23], t[24]);
        float best = (iou2 > iou1) ? 1.0f : 0.0f;

        float exists = t[20];
        float noobj  = 1.0f - exists;

        // pred_conf = (1-best)*p20 + best*p29   (reference's channel choice)
        float pred_conf = (1.0f - best) * p[20] + best * p[29];

        // object loss
        float d_obj = exists * pred_conf - exists * t[20];
        float obj_l = d_obj * d_obj;

        // no-object loss (both confidence channels vs target conf)
        float n1 = noobj * p[20] - noobj * t[20];
        float n2 = noobj * p[29] - noobj * t[20];
        float noobj_l = n1 * n1 + n2 * n2;

        // class loss over 20 channels
        float cls_l = 0.0f;
        #pragma unroll
        for (int k = 0; k < C_DIM; ++k) {
            float d = exists * p[k] - exists * t[k];
            cls_l += d * d;
        }

        // box_loss == 0 in the reference, LAMBDA_NOOBJ = 0.5
        acc += obj_l + 0.5f * noobj_l + cls_l;
    }

    __shared__ float sdata[THREADS];
    float bsum = block_reduce_256(acc, sdata);
    if (threadIdx.x == 0) partial[blockIdx.x] = bsum;
}

// ---------------------------------------------------------------------------
// Kernel 2: single-block deterministic final reduction of block partials.
// ---------------------------------------------------------------------------
__global__ void __launch_bounds__(THREADS)
yolo_loss_final(const float* __restrict__ partial, int n, float* __restrict__ out) {
    float acc = 0.0f;
    for (int i = threadIdx.x; i < n; i += THREADS) acc += partial[i];
    __shared__ float sdata[THREADS];
    float total = block_reduce_256(acc, sdata);
    if (threadIdx.x == 0) out[0] = total;
}

// ---------------------------------------------------------------------------
extern "C" void kernel_launch(void* const* d_in, const int* in_sizes, int n_in,
                              void* d_out, int out_size, void* d_ws, size_t ws_size,
                              hipStream_t stream) {
    const float* pred = (const float*)d_in[0];
    const float* tgt  = (const float*)d_in[1];
    float* out        = (float*)d_out;
    float* partial    = (float*)d_ws;

    int blocks = BLOCKS;
    if ((size_t)blocks * sizeof(float) > ws_size)
        blocks = (int)(ws_size / sizeof(float));   // ws is fixed per run -> still deterministic

    yolo_loss_partial<<<blocks, THREADS, 0, stream>>>(pred, tgt, partial, NCELLS);
    yolo_loss_final<<<1, THREADS, 0, stream>>>(partial, blocks, out);
}